// MeanCellExtrator_154618822672
// MI455X (gfx1250) — compile-verified
//
#include <hip/hip_runtime.h>

// Problem constants (match reference: B=4, C=16, H=W=1024, labels 1..512)
#define NB    4
#define NCH   16
#define MAXL  512
#define ENT   33                      // 16 pred sums + 16 target sums + 1 count
#define HW    (1024 * 1024)
#define BPI   128                     // blocks per image
#define TPB   256
#define VEC   4                       // pixels per thread per chunk (b128)
#define PIX_PER_BLOCK (HW / BPI)      // 8192
#define PIX_PER_CHUNK (TPB * VEC)     // 1024
#define NCHUNK (PIX_PER_BLOCK / PIX_PER_CHUNK) // 8
#define HIST   (MAXL * ENT)           // 16896 floats per histogram
#define WS_FLOATS (NB * HIST)         // 67584 floats (compact sums)
#define PART_FLOATS (NB * BPI * HIST) // 8,650,752 floats = 34.6MB (partials)
#define OUT_BLK (NB * MAXL * NCH)     // 32768 floats per means tensor

typedef int   v4i __attribute__((ext_vector_type(4)));
typedef float v4f __attribute__((ext_vector_type(4)));

// ---- CDNA5 async global->LDS path (ASYNCcnt-tracked) -----------------------
#if defined(__has_builtin)
#  if __has_builtin(__builtin_amdgcn_global_load_async_to_lds_b128) && \
      __has_builtin(__builtin_amdgcn_s_wait_asynccnt)
#    define HAVE_ASYNC_LDS 1
#  endif
#endif

typedef __attribute__((address_space(1))) v4i gv4i_t;
typedef __attribute__((address_space(3))) v4i lv4i_t;

__device__ __forceinline__ void async_label_copy4(const int* g, int* l) {
#ifdef HAVE_ASYNC_LDS
  // global_load_async_to_lds_b128: per-lane 16B global -> LDS, ASYNCcnt++
  __builtin_amdgcn_global_load_async_to_lds_b128((gv4i_t*)(g), (lv4i_t*)(l),
                                                 0, 0);
#else
  *(v4i*)l = *(const v4i*)g;
#endif
}

__device__ __forceinline__ void wait_async_le1() {
#ifdef HAVE_ASYNC_LDS
  __builtin_amdgcn_s_wait_asynccnt(1);
#endif
}
__device__ __forceinline__ void wait_async_le0() {
#ifdef HAVE_ASYNC_LDS
  __builtin_amdgcn_s_wait_asynccnt(0);
#endif
}

// ---- kernel 1: zero the compact accumulator (atomic-fallback path only) ----
__global__ void zero_ws_kernel(float* __restrict__ ws, int n) {
  int i = blockIdx.x * blockDim.x + threadIdx.x;
  if (i < n) ws[i] = 0.0f;
}

// ---- kernel 2: per-block LDS histogram -------------------------------------
__global__ __launch_bounds__(TPB) void accum_kernel(
    const float* __restrict__ pred, const float* __restrict__ target,
    const int* __restrict__ nuclei, float* __restrict__ ws, int use_partials) {
  // 512 labels * 33 entries * 4B = 67,584B of LDS (needs CDNA5's 320KB/WGP)
  __shared__ float hist[HIST];
  __shared__ __align__(16) int slab[2][PIX_PER_CHUNK]; // async label staging

  const int tid = threadIdx.x;
  const int b   = blockIdx.x / BPI;
  const int blk = blockIdx.x % BPI;

  const long long pixBase = (long long)blk * PIX_PER_BLOCK;
  const float* __restrict__ pb = pred   + (long long)b * NCH * HW;
  const float* __restrict__ tb = target + (long long)b * NCH * HW;
  const int*   __restrict__ nb = nuclei + (long long)b * HW;

  // kick off chunk-0 label fetch before zeroing LDS (overlaps with init)
  async_label_copy4(nb + pixBase + tid * VEC, &slab[0][tid * VEC]);

  for (int i = tid; i < HIST; i += TPB) hist[i] = 0.0f;
  __syncthreads();

  for (int ch = 0; ch < NCHUNK; ++ch) {
    const int cur = ch & 1;
    if (ch + 1 < NCHUNK) {
      // prefetch next chunk's labels; in-order completion => ASYNCcnt<=1
      // guarantees chunk `ch` has landed in LDS
      async_label_copy4(
          nb + pixBase + (long long)(ch + 1) * PIX_PER_CHUNK + tid * VEC,
          &slab[cur ^ 1][tid * VEC]);
      wait_async_le1();
    } else {
      wait_async_le0();
    }

    const v4i labs = *(const v4i*)&slab[cur][tid * VEC]; // own wave's slots
    const long long pix0 =
        pixBase + (long long)ch * PIX_PER_CHUNK + (long long)tid * VEC;

    // ---- pred phase (streaming b128 NT loads, 4 pixels x 16 channels) ----
    v4f pv[NCH];
#pragma unroll
    for (int c = 0; c < NCH; ++c)
      pv[c] = __builtin_nontemporal_load(
          (const v4f*)(pb + (long long)c * HW + pix0));
#pragma unroll
    for (int j = 0; j < VEC; ++j) {
      const int lab = labs[j];
      if (lab > 0) {                  // background (0) dropped by reference
        float* h = &hist[(lab - 1) * ENT];
#pragma unroll
        for (int c = 0; c < NCH; ++c) atomicAdd(&h[c], pv[c][j]); // ds_add_f32
        atomicAdd(&h[2 * NCH], 1.0f);                             // count
      }
    }

    // ---- target phase ----
    v4f tv[NCH];
#pragma unroll
    for (int c = 0; c < NCH; ++c)
      tv[c] = __builtin_nontemporal_load(
          (const v4f*)(tb + (long long)c * HW + pix0));
#pragma unroll
    for (int j = 0; j < VEC; ++j) {
      const int lab = labs[j];
      if (lab > 0) {
        float* h = &hist[(lab - 1) * ENT];
#pragma unroll
        for (int c = 0; c < NCH; ++c) atomicAdd(&h[NCH + c], tv[c][j]);
      }
    }
  }

  __syncthreads();
  if (use_partials) {
    // non-atomic flush: each block owns a private 16896-entry slice
    float* dst = ws + (long long)blockIdx.x * HIST;
    for (int i = tid; i < HIST; i += TPB)
      __builtin_nontemporal_store(hist[i], &dst[i]);
  } else {
    // fallback: atomic flush into compact per-image accumulator
    float* wsb = ws + (long long)b * HIST;
    for (int i = tid; i < HIST; i += TPB) {
      float v = hist[i];
      if (v != 0.0f) atomicAdd(&wsb[i], v);
    }
  }
}

// ---- kernel 3: sum the 128 per-block partials per image --------------------
__global__ void reduce_kernel(const float* __restrict__ part,
                              float* __restrict__ compact) {
  int i = blockIdx.x * blockDim.x + threadIdx.x;   // over NB*HIST entries
  if (i >= WS_FLOATS) return;
  const int b = i / HIST;
  const int e = i % HIST;
  const float* p = part + (long long)b * BPI * HIST + e;
  float s = 0.0f;
#pragma unroll 8
  for (int k = 0; k < BPI; ++k) s += p[(long long)k * HIST];
  compact[i] = s;
}

// ---- kernel 4: means + cell ids -------------------------------------------
__global__ void finalize_kernel(const float* __restrict__ ws,
                                float* __restrict__ out) {
  int i = blockIdx.x * blockDim.x + threadIdx.x;   // over B*MAXL*NCH
  if (i >= OUT_BLK) return;
  const int c   = i % NCH;
  const int row = i / NCH;                         // row = b*MAXL + s
  const int s   = row % MAXL;
  const float* e = ws + (long long)row * ENT;
  const float cnt   = e[2 * NCH];
  const float denom = fmaxf(cnt, 1.0f);
  out[row * NCH + c]           = e[c] / denom;           // pred_means
  out[OUT_BLK + row * NCH + c] = e[NCH + c] / denom;     // target_means
  if (c == 0)
    out[2 * OUT_BLK + row] = (cnt > 0.0f) ? (float)(s + 1) : 0.0f; // cell_ids
}

extern "C" void kernel_launch(void* const* d_in, const int* in_sizes, int n_in,
                              void* d_out, int out_size, void* d_ws,
                              size_t ws_size, hipStream_t stream) {
  const float* pred   = (const float*)d_in[0];
  const float* target = (const float*)d_in[1];
  const int*   nuclei = (const int*)d_in[2];
  float* ws  = (float*)d_ws;
  float* out = (float*)d_out;

  const size_t need_partials =
      ((size_t)PART_FLOATS + (size_t)WS_FLOATS) * sizeof(float);

  if (ws_size >= need_partials) {
    // fast path: private per-block partials + serial per-entry reduce (no global atomics)
    float* part    = ws;
    float* compact = ws + PART_FLOATS;
    accum_kernel<<<NB * BPI, TPB, 0, stream>>>(pred, target, nuclei, part, 1);
    reduce_kernel<<<(WS_FLOATS + 255) / 256, 256, 0, stream>>>(part, compact);
    finalize_kernel<<<(OUT_BLK + 255) / 256, 256, 0, stream>>>(compact, out);
  } else {
    // fallback: zero + atomic accumulation into compact workspace
    zero_ws_kernel<<<(WS_FLOATS + 255) / 256, 256, 0, stream>>>(ws, WS_FLOATS);
    accum_kernel<<<NB * BPI, TPB, 0, stream>>>(pred, target, nuclei, ws, 0);
    finalize_kernel<<<(OUT_BLK + 255) / 256, 256, 0, stream>>>(ws, out);
  }
}